// BasicBlock_4355096838467
// MI455X (gfx1250) — compile-verified
//
#include <hip/hip_runtime.h>
#include <hip/hip_bf16.h>

typedef float v2f __attribute__((ext_vector_type(2)));
typedef float v8f __attribute__((ext_vector_type(8)));

#define Nn   32
#define Cc   256
#define Hh   56
#define Wn   56
#define HW   (Hh * Wn)        // 3136
#define NHW  (Nn * HW)        // 100352  (GEMM N dimension)
#define KDIM (Cc * 9)         // 2304    (GEMM K dimension)
#define NELEM (Nn * Cc * HW)  // 25,690,112
#define BN_EPS 1e-5f

// ---------------------------------------------------------------------------
// One-shot weight reshape + mask fold:
//   Wrt[o][tap][ci] = W[o][ci][kh][kw] * mask[o],  tap = kh*3+kw.
// This makes the conv's A-fragment loads contiguous float2 in ci and deletes
// the per-element mask multiply from the conv epilogue (conv is linear).
// ---------------------------------------------------------------------------
__global__ __launch_bounds__(256)
void reshape_weights(const float* __restrict__ Wt, const float* __restrict__ mask,
                     float* __restrict__ Wrt)
{
  const int idx = blockIdx.x * 256 + threadIdx.x;   // over Cc*KDIM = 589,824
  if (idx >= Cc * KDIM) return;
  const int o   = idx / KDIM;
  const int r   = idx - o * KDIM;
  const int ci  = r / 9;
  const int tap = r - ci * 9;
  Wrt[(size_t)o * KDIM + tap * Cc + ci] = Wt[idx] * mask[o];
}

// ---------------------------------------------------------------------------
// Implicit-GEMM 3x3 conv (stride 1, pad 1), K-order = (tap, ci).
// Each wave computes a 64(M=Cout) x 16(N=NHW) output tile as 4 accumulators
// of V_WMMA_F32_16X16X4_F32, sharing one B fragment across the 4 M-tiles.
// Tap loop fully unrolled: bounds check + (dh,dw) math hoisted per tap;
// inner ci loop is pure pointer-stride adds. EXEC stays all-ones (branchless
// halo handling via clamped address * 0/1 scale).
// Block = 256 threads = 8 waves = 8 consecutive N-tiles.
// ---------------------------------------------------------------------------
__global__ __launch_bounds__(256)
void conv3x3_wmma_f32(const float* __restrict__ X,    // [N,C,H,W]
                      const float* __restrict__ Wrt,  // [O][tap][ci] (mask folded)
                      float* __restrict__ Y)          // [N,O,H,W]
{
  const int lane = threadIdx.x & 31;
  const int wave = threadIdx.x >> 5;
  const int half = lane >> 4;     // which 16-lane half of the wave
  const int lm   = lane & 15;     // A-row / B-col index for this lane

  const int ntile = blockIdx.x * 8 + wave;   // 0..6271 (NHW/16), exact tiling
  const int m0    = blockIdx.y * 64;         // gridDim.y = 4

  // This lane's GEMM column -> (n, h, w) of the output pixel
  const int col = ntile * 16 + lm;
  const int n   = col / HW;
  const int hw  = col - n * HW;
  const int h   = hw / Wn;
  const int w   = hw - h * Wn;

  // A rows for tile t: m = m0 + 16*t + lm ; this lane owns K = ...+half*2+{0,1}
  const float* A0 = Wrt + (size_t)(m0 + lm) * KDIM + half * 2;
  const float* Xn = X + (size_t)n * Cc * HW;   // + ci*HW + hh*Wn + ww

  v8f acc[4] = {v8f{}, v8f{}, v8f{}, v8f{}};

#pragma unroll
  for (int tap = 0; tap < 9; ++tap) {
    const int  dh    = tap / 3 - 1;
    const int  dw    = tap - (tap / 3) * 3 - 1;
    const int  hh    = h + dh;
    const int  ww    = w + dw;
    const bool valid = ((unsigned)hh < (unsigned)Hh) & ((unsigned)ww < (unsigned)Wn);
    const float bs   = valid ? 1.0f : 0.0f;           // halo -> contribute 0
    const float* Xb  = Xn + (size_t)(valid ? hh : 0) * Wn + (valid ? ww : 0)
                          + (size_t)(half * 2) * HW;  // this lane's first ci
    const float* Aw  = A0 + tap * Cc;

#pragma unroll 2
    for (int ci = 0; ci < Cc; ci += 4) {
      v2f b;
      b[0] = Xb[(size_t)ci * HW]      * bs;
      b[1] = Xb[(size_t)ci * HW + HW] * bs;

      const v2f a0 = *(const v2f*)(Aw + ci);
      const v2f a1 = *(const v2f*)(Aw + ci + 16 * KDIM);
      const v2f a2 = *(const v2f*)(Aw + ci + 32 * KDIM);
      const v2f a3 = *(const v2f*)(Aw + ci + 48 * KDIM);

      acc[0] = __builtin_amdgcn_wmma_f32_16x16x4_f32(false, a0, false, b, (short)0, acc[0], false, false);
      acc[1] = __builtin_amdgcn_wmma_f32_16x16x4_f32(false, a1, false, b, (short)0, acc[1], false, false);
      acc[2] = __builtin_amdgcn_wmma_f32_16x16x4_f32(false, a2, false, b, (short)0, acc[2], false, false);
      acc[3] = __builtin_amdgcn_wmma_f32_16x16x4_f32(false, a3, false, b, (short)0, acc[3], false, false);
    }
  }

  // D layout: VGPR v holds row m = (tile base) + v + 8*half, column = lane's col.
  float* Yc = Y + (size_t)n * Cc * HW + hw;
#pragma unroll
  for (int t = 0; t < 4; ++t) {
#pragma unroll
    for (int v = 0; v < 8; ++v) {
      const int co = m0 + t * 16 + v + 8 * half;
      Yc[(size_t)co * HW] = acc[t][v];
    }
  }
}

// ---------------------------------------------------------------------------
// Per-channel sum / sum-of-squares for training-mode BatchNorm.
// Grid: (C, N); one block reduces one (channel, image) plane of HW elements.
// ---------------------------------------------------------------------------
__global__ __launch_bounds__(256)
void bn_stats(const float* __restrict__ X, float* __restrict__ sum,
              float* __restrict__ sumsq)
{
  const int c = blockIdx.x;
  const int n = blockIdx.y;
  const float* p = X + (size_t)(n * Cc + c) * HW;

  float s = 0.0f, s2 = 0.0f;
  for (int i = threadIdx.x; i < HW; i += 256) {
    const float v = p[i];
    s += v;
    s2 += v * v;
  }

  __shared__ float sh[256], sh2[256];
  sh[threadIdx.x]  = s;
  sh2[threadIdx.x] = s2;
  __syncthreads();
  for (int off = 128; off > 0; off >>= 1) {
    if (threadIdx.x < off) {
      sh[threadIdx.x]  += sh[threadIdx.x + off];
      sh2[threadIdx.x] += sh2[threadIdx.x + off];
    }
    __syncthreads();
  }
  if (threadIdx.x == 0) {
    atomicAdd(&sum[c],   sh[0]);
    atomicAdd(&sumsq[c], sh2[0]);
  }
}

// ---------------------------------------------------------------------------
// y = relu( (x - mean) * rsqrt(var+eps) * gamma + beta ), in place.
// ---------------------------------------------------------------------------
__global__ __launch_bounds__(256)
void bn_relu_inplace(float* __restrict__ X, const float* __restrict__ g,
                     const float* __restrict__ b, const float* __restrict__ sum,
                     const float* __restrict__ sumsq)
{
  const size_t idx = (size_t)blockIdx.x * 256 + threadIdx.x;
  if (idx >= (size_t)NELEM) return;
  const int c = (int)((idx / HW) % Cc);
  const float inv_cnt = 1.0f / (float)NHW;
  const float mean = sum[c] * inv_cnt;
  const float var  = sumsq[c] * inv_cnt - mean * mean;
  const float inv  = rsqrtf(var + BN_EPS);
  const float y    = (X[idx] - mean) * inv * g[c] + b[c];
  X[idx] = y > 0.0f ? y : 0.0f;
}

// ---------------------------------------------------------------------------
// out = relu( bn2(out) + residual ), in place on `out`.
// ---------------------------------------------------------------------------
__global__ __launch_bounds__(256)
void bn_add_relu_inplace(float* __restrict__ Y, const float* __restrict__ Xres,
                         const float* __restrict__ g, const float* __restrict__ b,
                         const float* __restrict__ sum,
                         const float* __restrict__ sumsq)
{
  const size_t idx = (size_t)blockIdx.x * 256 + threadIdx.x;
  if (idx >= (size_t)NELEM) return;
  const int c = (int)((idx / HW) % Cc);
  const float inv_cnt = 1.0f / (float)NHW;
  const float mean = sum[c] * inv_cnt;
  const float var  = sumsq[c] * inv_cnt - mean * mean;
  const float inv  = rsqrtf(var + BN_EPS);
  const float y    = (Y[idx] - mean) * inv * g[c] + b[c] + Xres[idx];
  Y[idx] = y > 0.0f ? y : 0.0f;
}

extern "C" void kernel_launch(void* const* d_in, const int* in_sizes, int n_in,
                              void* d_out, int out_size, void* d_ws, size_t ws_size,
                              hipStream_t stream) {
  (void)in_sizes; (void)n_in; (void)out_size; (void)ws_size;

  const float* x      = (const float*)d_in[0];
  const float* W1     = (const float*)d_in[1];
  const float* W2     = (const float*)d_in[2];
  const float* gamma1 = (const float*)d_in[3];
  const float* beta1  = (const float*)d_in[4];
  const float* gamma2 = (const float*)d_in[5];
  const float* beta2  = (const float*)d_in[6];
  const float* mask1  = (const float*)d_in[7];
  const float* mask2  = (const float*)d_in[8];
  float* out = (float*)d_out;

  // Workspace: h buffer [NELEM] | stats [4*C] | reshaped weights [C*KDIM]
  float* hbuf  = (float*)d_ws;
  float* stats = hbuf + (size_t)NELEM;
  float* sum1  = stats;
  float* ss1   = stats + Cc;
  float* sum2  = stats + 2 * Cc;
  float* ss2   = stats + 3 * Cc;
  float* wrt   = stats + 4 * Cc;

  // Zero stat accumulators every call (d_ws is not re-poisoned between replays).
  hipMemsetAsync(stats, 0, 4 * Cc * sizeof(float), stream);

  const dim3 gConv(NHW / 16 / 8, Cc / 64);     // 784 x 4 blocks, 8 waves each
  const dim3 gStats(Cc, Nn);                   // 256 x 32
  const int  gElem = (NELEM + 255) / 256;
  const int  gWt   = (Cc * KDIM + 255) / 256;  // 2304 blocks

  // h = conv1(x) with mask1 folded into weights
  reshape_weights<<<gWt, 256, 0, stream>>>(W1, mask1, wrt);
  conv3x3_wmma_f32<<<gConv, 256, 0, stream>>>(x, wrt, hbuf);
  // h = relu(bn1(h))
  bn_stats<<<gStats, 256, 0, stream>>>(hbuf, sum1, ss1);
  bn_relu_inplace<<<gElem, 256, 0, stream>>>(hbuf, gamma1, beta1, sum1, ss1);
  // out_tmp = conv2(h) with mask2 folded into weights
  reshape_weights<<<gWt, 256, 0, stream>>>(W2, mask2, wrt);
  conv3x3_wmma_f32<<<gConv, 256, 0, stream>>>(hbuf, wrt, out);
  // out = relu(bn2(out_tmp) + x)
  bn_stats<<<gStats, 256, 0, stream>>>(out, sum2, ss2);
  bn_add_relu_inplace<<<gElem, 256, 0, stream>>>(out, x, gamma2, beta2, sum2, ss2);
}